// SparseGaussianHead_24352464569755
// MI455X (gfx1250) — compile-verified
//
#include <hip/hip_runtime.h>
#include <cmath>

typedef __attribute__((ext_vector_type(2))) float v2f;
typedef __attribute__((ext_vector_type(8))) float v8f;

#define NPTS   400000
#define CIN    64
#define COUT   38
#define KOFF   27
#define HSTRIDE 40          // padded h row stride (38 -> 40)
#define LDS1_STRIDE 68      // 64 + 4 pad -> conflict-free float2 DS reads
#define LDS2_STRIDE 44      // 40 + 4 pad

// packed transposed weight tile sizes (in floats)
#define W1T_TILE (48 * 64)   // 3072
#define W2T_TILE (48 * 40)   // 1920

// ws layout (floats):
#define WS_ZROW  0                            // 64 zero floats (layer-1 padding row)
#define WS_W1T   64                           // 27 * 3072
#define WS_W2T   (WS_W1T + KOFF * W1T_TILE)   // 27 * 1920
#define WS_HBUF  (WS_W2T + KOFF * W2T_TILE)   // (N+1) * HSTRIDE

__device__ __forceinline__ float gelu_exact(float x) {
    return 0.5f * x * (1.0f + erff(x * 0.7071067811865476f));
}

// One-time prep: zero rows + transpose/zero-pad W1, W2 into packed tiles.
// grid: 324 blocks x 256 = 82944 threads == 27*3072 exactly.
__global__ void sgh_prep(const float* __restrict__ W1,
                         const float* __restrict__ W2,
                         float* __restrict__ ws)
{
    const int g = blockIdx.x * 256 + threadIdx.x;

    if (g < 64) ws[WS_ZROW + g] = 0.0f;
    if (g < HSTRIDE) ws[WS_HBUF + (size_t)NPTS * HSTRIDE + g] = 0.0f;

    { // W1t[k][n*64 + c] = W1[k][c][n] (n >= 38 -> 0)
        int k = g / W1T_TILE, r = g % W1T_TILE;
        int n = r >> 6, c = r & 63;
        float v = (n < COUT) ? W1[((size_t)k * CIN + c) * COUT + n] : 0.0f;
        ws[WS_W1T + g] = v;
    }
    if (g < KOFF * W2T_TILE) { // W2t[k][n*40 + c] = W2[k][c][n] (pads -> 0)
        int k = g / W2T_TILE, r = g % W2T_TILE;
        int n = r / HSTRIDE, c = r % HSTRIDE;
        float v = (n < COUT && c < COUT) ? W2[((size_t)k * COUT + c) * COUT + n] : 0.0f;
        ws[WS_W2T + g] = v;
    }
}

// Layer 1: h = gelu(sum_k feat[nbr[k]] @ W1[k]).  Two 16-row M tiles per wave.
__global__ __launch_bounds__(256)
void sgh_conv1(const float* __restrict__ feat,
               const int*   __restrict__ nbr,
               float*       __restrict__ ws)
{
    __shared__ float Wt[48 * LDS1_STRIDE];

    const float* zrow = ws + WS_ZROW;
    const float* W1t  = ws + WS_W1T;
    float*       hbuf = ws + WS_HBUF;

    const int tid  = threadIdx.x;
    const int lane = tid & 31;
    const int wave = tid >> 5;
    const int half = lane >> 4;
    const int mrow = lane & 15;
    const int ncol = lane & 15;

    const int m0 = (blockIdx.x * 8 + wave) * 32;   // 32 rows per wave (2 tiles)
    const int iA = m0 + mrow;
    const int iB = m0 + 16 + mrow;
    const int icA = (iA < NPTS) ? iA : (NPTS - 1);
    const int icB = (iB < NPTS) ? iB : (NPTS - 1);

    v8f a0 = {}, a1 = {}, a2 = {};   // tile A accumulators (cols 0..15,16..31,32..47)
    v8f b0 = {}, b1 = {}, b2 = {};   // tile B accumulators

    for (int k = 0; k < KOFF; ++k) {
        __syncthreads();
        { // contiguous coalesced copy of pre-transposed tile into padded LDS
            const float* Wg = W1t + (size_t)k * W1T_TILE;
            #pragma unroll
            for (int it = 0; it < W1T_TILE / 256; ++it) {
                int t = tid + it * 256;
                Wt[(t >> 6) * LDS1_STRIDE + (t & 63)] = Wg[t];
            }
        }
        __syncthreads();

        const int idxA = nbr[(size_t)k * NPTS + icA];
        const int idxB = nbr[(size_t)k * NPTS + icB];
        const float* arowA = (iA < NPTS && idxA < NPTS) ? (feat + (size_t)idxA * CIN) : zrow;
        const float* arowB = (iB < NPTS && idxB < NPTS) ? (feat + (size_t)idxB * CIN) : zrow;

        #pragma unroll
        for (int kk = 0; kk < CIN / 4; ++kk) {
            v2f fA = *(const v2f*)(arowA + 4 * kk + 2 * half);
            v2f fB = *(const v2f*)(arowB + 4 * kk + 2 * half);
            const float* bb = &Wt[ncol * LDS1_STRIDE + 4 * kk + 2 * half];
            v2f w0 = *(const v2f*)(bb);
            v2f w1 = *(const v2f*)(bb + 16 * LDS1_STRIDE);
            v2f w2 = *(const v2f*)(bb + 32 * LDS1_STRIDE);
            a0 = __builtin_amdgcn_wmma_f32_16x16x4_f32(false, fA, false, w0, (short)0, a0, false, false);
            a1 = __builtin_amdgcn_wmma_f32_16x16x4_f32(false, fA, false, w1, (short)0, a1, false, false);
            a2 = __builtin_amdgcn_wmma_f32_16x16x4_f32(false, fA, false, w2, (short)0, a2, false, false);
            b0 = __builtin_amdgcn_wmma_f32_16x16x4_f32(false, fB, false, w0, (short)0, b0, false, false);
            b1 = __builtin_amdgcn_wmma_f32_16x16x4_f32(false, fB, false, w1, (short)0, b1, false, false);
            b2 = __builtin_amdgcn_wmma_f32_16x16x4_f32(false, fB, false, w2, (short)0, b2, false, false);
        }
    }

    #pragma unroll
    for (int r = 0; r < 8; ++r) {
        int rowA = m0 + r + 8 * half;
        if (rowA < NPTS) {
            float* hr = hbuf + (size_t)rowA * HSTRIDE;
            hr[ncol]      = gelu_exact(a0[r]);
            hr[16 + ncol] = gelu_exact(a1[r]);
            if (ncol < HSTRIDE - 32) hr[32 + ncol] = gelu_exact(a2[r]);
        }
        int rowB = m0 + 16 + r + 8 * half;
        if (rowB < NPTS) {
            float* hr = hbuf + (size_t)rowB * HSTRIDE;
            hr[ncol]      = gelu_exact(b0[r]);
            hr[16 + ncol] = gelu_exact(b1[r]);
            if (ncol < HSTRIDE - 32) hr[32 + ncol] = gelu_exact(b2[r]);
        }
    }
}

// Layer 2: out = sum_k h[nbr[k]] @ W2[k] (K padded to 40; pad rows of W are 0)
__global__ __launch_bounds__(256)
void sgh_conv2(const int*   __restrict__ nbr,
               const float* __restrict__ ws,
               float*       __restrict__ out)
{
    __shared__ float Wt[48 * LDS2_STRIDE];

    const float* W2t  = ws + WS_W2T;
    const float* hbuf = ws + WS_HBUF;

    const int tid  = threadIdx.x;
    const int lane = tid & 31;
    const int wave = tid >> 5;
    const int half = lane >> 4;
    const int mrow = lane & 15;
    const int ncol = lane & 15;

    const int m0 = (blockIdx.x * 8 + wave) * 32;
    const int iA = m0 + mrow;
    const int iB = m0 + 16 + mrow;
    const int icA = (iA < NPTS) ? iA : (NPTS - 1);
    const int icB = (iB < NPTS) ? iB : (NPTS - 1);

    v8f a0 = {}, a1 = {}, a2 = {};
    v8f b0 = {}, b1 = {}, b2 = {};

    for (int k = 0; k < KOFF; ++k) {
        __syncthreads();
        { // 1920 floats: 7 full rounds of 256 + guarded partial
            const float* Wg = W2t + (size_t)k * W2T_TILE;
            #pragma unroll
            for (int it = 0; it < 8; ++it) {
                int t = tid + it * 256;
                if (t < W2T_TILE)
                    Wt[(t / HSTRIDE) * LDS2_STRIDE + (t % HSTRIDE)] = Wg[t];
            }
        }
        __syncthreads();

        int idxA = nbr[(size_t)k * NPTS + icA];
        int idxB = nbr[(size_t)k * NPTS + icB];
        if (iA >= NPTS || idxA > NPTS) idxA = NPTS;   // zero row of hbuf
        if (iB >= NPTS || idxB > NPTS) idxB = NPTS;
        const float* arowA = hbuf + (size_t)idxA * HSTRIDE;
        const float* arowB = hbuf + (size_t)idxB * HSTRIDE;

        #pragma unroll
        for (int kk = 0; kk < HSTRIDE / 4; ++kk) {
            v2f fA = *(const v2f*)(arowA + 4 * kk + 2 * half);
            v2f fB = *(const v2f*)(arowB + 4 * kk + 2 * half);
            const float* bb = &Wt[ncol * LDS2_STRIDE + 4 * kk + 2 * half];
            v2f w0 = *(const v2f*)(bb);
            v2f w1 = *(const v2f*)(bb + 16 * LDS2_STRIDE);
            v2f w2 = *(const v2f*)(bb + 32 * LDS2_STRIDE);
            a0 = __builtin_amdgcn_wmma_f32_16x16x4_f32(false, fA, false, w0, (short)0, a0, false, false);
            a1 = __builtin_amdgcn_wmma_f32_16x16x4_f32(false, fA, false, w1, (short)0, a1, false, false);
            a2 = __builtin_amdgcn_wmma_f32_16x16x4_f32(false, fA, false, w2, (short)0, a2, false, false);
            b0 = __builtin_amdgcn_wmma_f32_16x16x4_f32(false, fB, false, w0, (short)0, b0, false, false);
            b1 = __builtin_amdgcn_wmma_f32_16x16x4_f32(false, fB, false, w1, (short)0, b1, false, false);
            b2 = __builtin_amdgcn_wmma_f32_16x16x4_f32(false, fB, false, w2, (short)0, b2, false, false);
        }
    }

    #pragma unroll
    for (int r = 0; r < 8; ++r) {
        int rowA = m0 + r + 8 * half;
        if (rowA < NPTS) {
            float* orow = out + (size_t)rowA * COUT;
            orow[ncol]      = a0[r];
            orow[16 + ncol] = a1[r];
            if (ncol < COUT - 32) orow[32 + ncol] = a2[r];
        }
        int rowB = m0 + 16 + r + 8 * half;
        if (rowB < NPTS) {
            float* orow = out + (size_t)rowB * COUT;
            orow[ncol]      = b0[r];
            orow[16 + ncol] = b1[r];
            if (ncol < COUT - 32) orow[32 + ncol] = b2[r];
        }
    }
}

extern "C" void kernel_launch(void* const* d_in, const int* in_sizes, int n_in,
                              void* d_out, int out_size, void* d_ws, size_t ws_size,
                              hipStream_t stream) {
    const float* feat = (const float*)d_in[0];   // [N,64] f32
    const int*   nbr  = (const int*)  d_in[1];   // [27,N] i32
    const float* W1   = (const float*)d_in[2];   // [27,64,38] f32
    const float* W2   = (const float*)d_in[3];   // [27,38,38] f32
    float* out = (float*)d_out;                  // [N,38] f32
    float* ws  = (float*)d_ws;

    sgh_prep<<<(KOFF * W1T_TILE) / 256, 256, 0, stream>>>(W1, W2, ws);

    const int blocks = (NPTS + 255) / 256;       // 8 waves/block, 32 rows/wave
    sgh_conv1<<<blocks, 256, 0, stream>>>(feat, nbr, ws);
    sgh_conv2<<<blocks, 256, 0, stream>>>(nbr, ws, out);
}